// FFT_for_Decomp_45079976739080
// MI455X (gfx1250) — compile-verified
//
#include <hip/hip_runtime.h>
#include <hip/hip_bf16.h>
#include <math.h>

// ---------------------------------------------------------------------------
// Problem constants
// ---------------------------------------------------------------------------
constexpr int B_  = 32;    // batch
constexpr int L_  = 336;   // sequence length (== irfft n)
constexpr int C_  = 512;   // channels
constexpr int T_  = 169;   // rfft frequencies (L/2+1)
constexpr int TP  = 176;   // T_ padded to multiple of 16
constexpr int FPAD = 340;  // DFT table row pitch (336 + 4) -> LDS/bank friendly, even
constexpr int GPAD = 180;  // inverse table row pitch (176 + 4), even

constexpr int XROW = 516;  // LDS row pitch for X in stage 2 (512+4 -> conflict free)

typedef float v2f __attribute__((ext_vector_type(2)));
typedef float v8f __attribute__((ext_vector_type(8)));

// D = A(16x4 f32) * B(4x16 f32) + C(16x16 f32)   -- CDNA5 wave32 WMMA
__device__ __forceinline__ v8f wmma4(v2f a, v2f b, v8f c) {
  return __builtin_amdgcn_wmma_f32_16x16x4_f32(
      /*neg_a=*/false, a, /*neg_b=*/false, b,
      /*c_mod=*/(short)0, c, /*reuse_a=*/false, /*reuse_b=*/false);
}

// ---------------------------------------------------------------------------
// Kernel 1: coef[i] = softmax(weights)[i] * scale_i / sqrt(n); pad to TP
// ---------------------------------------------------------------------------
__global__ __launch_bounds__(256) void coef_kernel(const float* __restrict__ w,
                                                   float* __restrict__ coef) {
  __shared__ float red[256];
  const int tid = threadIdx.x;

  float m = -1e30f;
  for (int i = tid; i < T_; i += 256) m = fmaxf(m, w[i]);
  red[tid] = m;
  __syncthreads();
  for (int s = 128; s > 0; s >>= 1) {
    if (tid < s) red[tid] = fmaxf(red[tid], red[tid + s]);
    __syncthreads();
  }
  const float mx = red[0];
  __syncthreads();

  float sum = 0.f;
  for (int i = tid; i < T_; i += 256) sum += expf(w[i] - mx);
  red[tid] = sum;
  __syncthreads();
  for (int s = 128; s > 0; s >>= 1) {
    if (tid < s) red[tid] += red[tid + s];
    __syncthreads();
  }
  const float denom  = red[0];
  const float invsqn = rsqrtf((float)L_);

  for (int i = tid; i < TP; i += 256) {
    float cv = 0.f;
    if (i < T_) {
      const float scale = (i == 0 || i == T_ - 1) ? 1.f : 2.f;
      cv = (expf(w[i] - mx) / denom) * scale * invsqn;
    }
    coef[i] = cv;
  }
}

// ---------------------------------------------------------------------------
// Kernel 2: trig tables.
//   Fc[k][l] =  cos(2*pi*k*l/L)/sqrt(L)   (k<T_, l<L_, else 0)   [TP x FPAD]
//   Fs[k][l] = -sin(2*pi*k*l/L)/sqrt(L)                           [TP x FPAD]
//   Gc[l][i] =  cos(2*pi*i*l/L)           (i<T_, else 0)          [L_ x GPAD]
//   Gs[l][i] = -sin(2*pi*i*l/L)                                   [L_ x GPAD]
// Zero padding kills contributions from padded K rows (incl. poisoned y rows).
// ---------------------------------------------------------------------------
__global__ __launch_bounds__(256) void tables_kernel(float* __restrict__ Fc,
                                                     float* __restrict__ Fs,
                                                     float* __restrict__ Gc,
                                                     float* __restrict__ Gs) {
  const int idx = blockIdx.x * 256 + threadIdx.x;
  const float TWO_PI = 6.283185307179586f;
  const float invL   = 1.0f / (float)L_;

  if (idx < TP * FPAD) {
    const int k = idx / FPAD, l = idx % FPAD;
    float vc = 0.f, vs = 0.f;
    if (k < T_ && l < L_) {
      const float th = TWO_PI * (float)((k * l) % L_) * invL;
      const float s  = rsqrtf((float)L_);
      vc =  cosf(th) * s;
      vs = -sinf(th) * s;
    }
    Fc[idx] = vc;
    Fs[idx] = vs;
  }
  if (idx < L_ * GPAD) {
    const int l = idx / GPAD, i = idx % GPAD;
    float vc = 0.f, vs = 0.f;
    if (i < T_) {
      const float th = TWO_PI * (float)((i * l) % L_) * invL;
      vc =  cosf(th);
      vs = -sinf(th);
    }
    Gc[idx] = vc;
    Gs[idx] = vs;
  }
}

// ---------------------------------------------------------------------------
// Kernel 3: stage-1 rfft as GEMM.
//   xfr[b,k,c] = sum_l Fc[k,l] * x[b,l,c]   (M=k tile 16, K=l=336, N=c)
//   xfi[b,k,c] = sum_l Fs[k,l] * x[b,l,c]
// grid = (4 n-blocks of 128, 11 k-tiles, 32 batches), block = 256 (8 waves)
// ---------------------------------------------------------------------------
__global__ __launch_bounds__(256) void rfft_kernel(const float* __restrict__ x,
                                                   const float* __restrict__ Fc,
                                                   const float* __restrict__ Fs,
                                                   float* __restrict__ xfr,
                                                   float* __restrict__ xfi) {
  const int tid   = threadIdx.x;
  const int wave  = tid >> 5;
  const int lane  = tid & 31;
  const int lhalf = lane >> 4;
  const int l16   = lane & 15;

  const int c  = blockIdx.x * 128 + wave * 16 + l16;  // output column
  const int k0 = blockIdx.y * 16;                     // k-tile base
  const int b  = blockIdx.z;

  const float* xb = x + (size_t)b * (L_ * C_);

  v8f accR = {};
  v8f accI = {};

  for (int l0 = 0; l0 < L_; l0 += 4) {
    const int la = l0 + 2 * lhalf;
    // A frags (16x4): VGPR0 <- K=2*lhalf, VGPR1 <- K=2*lhalf+1, row M=l16
    const v2f ac = *(const v2f*)(Fc + (size_t)(k0 + l16) * FPAD + la);
    const v2f as = *(const v2f*)(Fs + (size_t)(k0 + l16) * FPAD + la);
    // B frag (4x16): VGPR0 <- row K=2*lhalf, VGPR1 <- row K=2*lhalf+1, col N=l16
    v2f bx;
    bx.x = xb[(size_t)la * C_ + c];
    bx.y = xb[(size_t)(la + 1) * C_ + c];
    accR = wmma4(ac, bx, accR);
    accI = wmma4(as, bx, accI);
  }

  // C/D layout: VGPR r, lanes0-15 -> M=r, lanes16-31 -> M=r+8; N = l16
#pragma unroll
  for (int r = 0; r < 8; ++r) {
    const int k = k0 + r + 8 * lhalf;
    const size_t off = ((size_t)b * TP + k) * C_ + c;
    xfr[off] = accR[r];
    xfi[off] = accI[r];
  }
}

// ---------------------------------------------------------------------------
// Kernel 4 (dominant): per-frequency complex GEMM, coef+bias fused epilogue.
//   yr[b,i,o] = coef[i] * ( sum_c xr*Wr - xi*Wi + br[i,o] )
//   yi[b,i,o] = coef[i] * ( sum_c xr*Wi + xi*Wr + bi[i,o] )
// One block = one frequency x 128 output columns; all 32 batch rows.
// X (32x512 real + imag) staged in LDS with 516-float pitch (conflict-free
// ds_load_b64 A-frags). Weights stream from HBM exactly once as B-frags.
// grid = (4, 169), block = 256 (8 waves), dynamic LDS = 2*32*516*4 B.
// ---------------------------------------------------------------------------
__global__ __launch_bounds__(256) void freqmm_kernel(const float* __restrict__ Wr,
                                                     const float* __restrict__ Wi,
                                                     const float* __restrict__ br,
                                                     const float* __restrict__ bi,
                                                     const float* __restrict__ coef,
                                                     const float* __restrict__ xfr,
                                                     const float* __restrict__ xfi,
                                                     float* __restrict__ yr,
                                                     float* __restrict__ yi) {
  extern __shared__ float lds[];  // [2][32][XROW]
  const int tid = threadIdx.x;
  const int i   = blockIdx.y;  // frequency

  // ---- cooperative X load: 2 matrices * 32 rows * 512 floats = 8192 float4
  for (int j = 0; j < 32; ++j) {
    const int idx = tid + 256 * j;   // float4 index
    const int m   = idx >> 12;       // 0 = real, 1 = imag
    const int r   = idx & 4095;
    const int bb  = r >> 7;          // batch row 0..31
    const int c4  = (r & 127) << 2;  // column
    const float* src = (m ? xfi : xfr) + ((size_t)bb * TP + i) * C_ + c4;
    const float4 v = *(const float4*)src;
    *(float4*)(lds + (size_t)m * (32 * XROW) + bb * XROW + c4) = v;
  }
  __syncthreads();

  const int wave  = tid >> 5;
  const int lane  = tid & 31;
  const int lhalf = lane >> 4;
  const int l16   = lane & 15;
  const int o     = blockIdx.x * 128 + wave * 16 + l16;  // output channel

  const float* wr_row = Wr + ((size_t)i * C_ + o) * C_;
  const float* wi_row = Wi + ((size_t)i * C_ + o) * C_;

  v8f aR0 = {}, aR1 = {}, aI0 = {}, aI1 = {};

  for (int kk = 0; kk < C_; kk += 4) {
    if ((kk & 127) == 0) {
      __builtin_prefetch(wr_row + kk + 256, 0, 0);
      __builtin_prefetch(wi_row + kk + 256, 0, 0);
    }
    const int cidx = kk + 2 * lhalf;
    // A frags from LDS (M = batch row, K = c)
    const v2f xr0 = *(const v2f*)(lds + (size_t)l16 * XROW + cidx);
    const v2f xr1 = *(const v2f*)(lds + (size_t)(l16 + 16) * XROW + cidx);
    const v2f xi0 = *(const v2f*)(lds + 32 * XROW + (size_t)l16 * XROW + cidx);
    const v2f xi1 = *(const v2f*)(lds + 32 * XROW + (size_t)(l16 + 16) * XROW + cidx);
    // B frags from global weights (K = c, N = o)
    const v2f bwr = *(const v2f*)(wr_row + cidx);
    const v2f bwi = *(const v2f*)(wi_row + cidx);
    // f32 WMMA has no A/B negate modifier -> negate imag A in VALU
    const v2f nxi0 = -xi0, nxi1 = -xi1;

    aR0 = wmma4(xr0,  bwr, aR0);
    aR0 = wmma4(nxi0, bwi, aR0);
    aR1 = wmma4(xr1,  bwr, aR1);
    aR1 = wmma4(nxi1, bwi, aR1);
    aI0 = wmma4(xr0,  bwi, aI0);
    aI0 = wmma4(xi0,  bwr, aI0);
    aI1 = wmma4(xr1,  bwi, aI1);
    aI1 = wmma4(xi1,  bwr, aI1);
  }

  const float cf = coef[i];
  const float bR = br[(size_t)i * C_ + o];
  const float bI = bi[(size_t)i * C_ + o];

#pragma unroll
  for (int r = 0; r < 8; ++r) {
    const int m = r + 8 * lhalf;  // batch row within tile
    const size_t off0 = ((size_t)m * TP + i) * C_ + o;
    const size_t off1 = ((size_t)(m + 16) * TP + i) * C_ + o;
    yr[off0] = cf * (aR0[r] + bR);
    yi[off0] = cf * (aI0[r] + bI);
    yr[off1] = cf * (aR1[r] + bR);
    yi[off1] = cf * (aI1[r] + bI);
  }
}

// ---------------------------------------------------------------------------
// Kernel 5: stage-3 inverse transform + weighted sum as GEMM over i.
//   out[b,l,c] = sum_i Gc[l,i]*yr[b,i,c] + Gs[l,i]*yi[b,i,c]
// (coef already folded into yr/yi; Gs holds -sin; padded i-cols are zero.)
// grid = (4 n-blocks, 21 l-tiles, 32 batches), block = 256 (8 waves)
// ---------------------------------------------------------------------------
__global__ __launch_bounds__(256) void irfft_kernel(const float* __restrict__ Gc,
                                                    const float* __restrict__ Gs,
                                                    const float* __restrict__ yr,
                                                    const float* __restrict__ yi,
                                                    float* __restrict__ out) {
  const int tid   = threadIdx.x;
  const int wave  = tid >> 5;
  const int lane  = tid & 31;
  const int lhalf = lane >> 4;
  const int l16   = lane & 15;

  const int c  = blockIdx.x * 128 + wave * 16 + l16;
  const int l0 = blockIdx.y * 16;
  const int b  = blockIdx.z;

  const float* yrb = yr + (size_t)b * TP * C_;
  const float* yib = yi + (size_t)b * TP * C_;

  v8f acc = {};

  for (int i0 = 0; i0 < TP; i0 += 4) {
    const int ia = i0 + 2 * lhalf;
    const v2f ac = *(const v2f*)(Gc + (size_t)(l0 + l16) * GPAD + ia);
    const v2f as = *(const v2f*)(Gs + (size_t)(l0 + l16) * GPAD + ia);
    v2f bre, bim;
    bre.x = yrb[(size_t)ia * C_ + c];
    bre.y = yrb[(size_t)(ia + 1) * C_ + c];
    bim.x = yib[(size_t)ia * C_ + c];
    bim.y = yib[(size_t)(ia + 1) * C_ + c];
    acc = wmma4(ac, bre, acc);
    acc = wmma4(as, bim, acc);
  }

#pragma unroll
  for (int r = 0; r < 8; ++r) {
    const int l = l0 + r + 8 * lhalf;
    out[((size_t)b * L_ + l) * C_ + c] = acc[r];
  }
}

// ---------------------------------------------------------------------------
// Launch
// ---------------------------------------------------------------------------
extern "C" void kernel_launch(void* const* d_in, const int* in_sizes, int n_in,
                              void* d_out, int out_size, void* d_ws, size_t ws_size,
                              hipStream_t stream) {
  const float* x  = (const float*)d_in[0];  // [32,336,512]
  const float* Wr = (const float*)d_in[1];  // [169,512,512]
  const float* Wi = (const float*)d_in[2];  // [169,512,512]
  const float* br = (const float*)d_in[3];  // [169,512]
  const float* bi = (const float*)d_in[4];  // [169,512]
  const float* w  = (const float*)d_in[5];  // [169]
  float* out = (float*)d_out;               // [32,336,512]

  float* ws   = (float*)d_ws;
  float* coef = ws;                         // 176
  float* Fc   = coef + TP;                  // TP*FPAD
  float* Fs   = Fc + (size_t)TP * FPAD;     // TP*FPAD
  float* Gc   = Fs + (size_t)TP * FPAD;     // L_*GPAD
  float* Gs   = Gc + (size_t)L_ * GPAD;     // L_*GPAD
  float* xfr  = Gs + (size_t)L_ * GPAD;     // B_*TP*C_
  float* xfi  = xfr + (size_t)B_ * TP * C_;
  float* yrw  = xfi + (size_t)B_ * TP * C_;
  float* yiw  = yrw + (size_t)B_ * TP * C_;

  coef_kernel<<<1, 256, 0, stream>>>(w, coef);

  const int ntabF = TP * FPAD;              // 59,840
  const int ntabG = L_ * GPAD;              // 60,480
  const int ntab  = (ntabF > ntabG) ? ntabF : ntabG;
  tables_kernel<<<(ntab + 255) / 256, 256, 0, stream>>>(Fc, Fs, Gc, Gs);

  rfft_kernel<<<dim3(4, TP / 16, B_), 256, 0, stream>>>(x, Fc, Fs, xfr, xfi);

  const size_t ldsBytes = (size_t)2 * 32 * XROW * sizeof(float);  // 132,096 B
  freqmm_kernel<<<dim3(4, T_, 1), 256, ldsBytes, stream>>>(Wr, Wi, br, bi, coef,
                                                           xfr, xfi, yrw, yiw);

  irfft_kernel<<<dim3(4, L_ / 16, B_), 256, 0, stream>>>(Gc, Gs, yrw, yiw, out);
}